// Serialization_53171695125268
// MI455X (gfx1250) — compile-verified
//
#include <hip/hip_runtime.h>
#include <hip/hip_bf16.h>
#include <stdint.h>

// Problem sizes (from reference setup_inputs)
#define BB 16
#define NN 8192
#define DD 256
#define NDEPTH 16
// scales = 512 >> i  ==> zero for i >= 10: steps 10..15 cannot change `total`.
#define NACTIVE 10
#define T1 1024
#define PPT (NN / T1)          // 8 points per thread
#define ROWS_PER_DESC 16       // TDM gather: up to 16 x 16-bit row indices per descriptor
#define WAVES2 2
#define T2 (WAVES2 * 32)
#define NGROUP (BB * NN / ROWS_PER_DESC)   // 8192 wave-tasks for the gather

typedef float v2f __attribute__((ext_vector_type(2)));
typedef float v8f __attribute__((ext_vector_type(8)));
typedef unsigned int v4u __attribute__((ext_vector_type(4)));
typedef int v4i __attribute__((ext_vector_type(4)));
typedef int v8i __attribute__((ext_vector_type(8)));

#if defined(__HIP_DEVICE_COMPILE__) && __has_builtin(__builtin_amdgcn_wmma_f32_16x16x4_f32)
#define USE_WMMA44 1
#endif
#if defined(__HIP_DEVICE_COMPILE__) && \
    __has_builtin(__builtin_amdgcn_tensor_load_to_lds) && \
    __has_builtin(__builtin_amdgcn_tensor_store_from_lds) && \
    __has_builtin(__builtin_amdgcn_s_wait_tensorcnt)
#define USE_TDM 1
#endif

// ---------------- RNG (counter-hash; JAX threefry is not bit-reproducible) ---
__device__ __forceinline__ unsigned hash_u32(unsigned x) {
  x ^= x >> 16; x *= 0x7feb352dU; x ^= x >> 15; x *= 0x846ca68bU; x ^= x >> 16;
  return x;
}
__device__ __forceinline__ float u01(unsigned h) {
  return (float)(h >> 8) * (1.0f / 16777216.0f);
}
__device__ __forceinline__ float gaussf(unsigned s) {
  float u1 = fmaxf(u01(hash_u32(s * 2654435761u + 0x9E3779B9u)), 1e-7f);
  float u2 = u01(hash_u32(s ^ 0x85EBCA6Bu));
  return sqrtf(-2.0f * __logf(u1)) * __cosf(6.28318530718f * u2);
}

// ---------------- block-wide reductions (1024 threads) ----------------------
__device__ __forceinline__ float block_sum(float v, float* redf, int tid) {
  redf[tid] = v; __syncthreads();
  for (int s = T1 >> 1; s > 0; s >>= 1) {
    if (tid < s) redf[tid] += redf[tid + s];
    __syncthreads();
  }
  float r = redf[0]; __syncthreads();
  return r;
}
__device__ __forceinline__ unsigned long long
block_min(unsigned long long v, unsigned long long* red, int tid) {
  red[tid] = v; __syncthreads();
  for (int s = T1 >> 1; s > 0; s >>= 1) {
    if (tid < s) { unsigned long long o = red[tid + s]; if (o < red[tid]) red[tid] = o; }
    __syncthreads();
  }
  unsigned long long r = red[0]; __syncthreads();
  return r;
}

// ============================================================================
// Kernel 1: serialization. One 1024-thread workgroup per batch.
// Points live in registers (8/thread); LDS for reductions + bitonic sort.
// The per-step projection (the op's only matmul) runs on V_WMMA_F32_16X16X4_F32.
// ============================================================================
__global__ __launch_bounds__(T1)
void serialize_kernel(const float* __restrict__ coords,
                      const float* __restrict__ mask,
                      int* __restrict__ idx_ws,
                      float* __restrict__ out) {
  __shared__ int comp[NN];                     // 32 KB sort keys
  __shared__ unsigned long long redll[T1];     // 8 KB argmin reduction
  __shared__ float redf[T1];                   // 4 KB sum reduction
  __shared__ float dup[4][3];                  // duplet points

  const int b = blockIdx.x;
  const int tid = threadIdx.x;
  const int lane = tid & 31;
  const bool lolane = lane < 16;

  float px[PPT], py[PPT], pz[PPT], mk[PPT], ml[PPT], mr[PPT], projv[PPT];
  int tot[PPT];

#pragma unroll
  for (int k = 0; k < PPT; ++k) {
    int i = tid + k * T1;
    size_t base = (size_t)b * NN + i;
    px[k] = coords[base * 3 + 0];
    py[k] = coords[base * 3 + 1];
    pz[k] = coords[base * 3 + 2];
    mk[k] = mask[base];
    tot[k] = 0;
  }

  float part = 0.f;
#pragma unroll
  for (int k = 0; k < PPT; ++k) part += mk[k];
  const float nsum = block_sum(part, redf, tid);

  float oax = 1.f, oay = 1.f, oaz = 1.f;   // oldABC = ones

  for (int d = 0; d < NACTIVE; ++d) {
    const int scale = 512 >> d;
    const unsigned sd = ((unsigned)b * 0x9E3779B1u) ^ ((unsigned)d * 0x85EBCA77u);
    float ax = gaussf(sd * 3u + 11u);
    float ay = gaussf(sd * 3u + 12u);
    float az = gaussf(sd * 3u + 13u);
    // sign(dot(normalize(ABC), normalize(old))) == sign(dot(ABC, old)); sign(0)->+1
    float ang = ax * oax + ay * oay + az * oaz;
    float sg = (ang < 0.f) ? -1.f : 1.f;
    ax *= sg; ay *= sg; az *= sg;
    oax = ax; oay = ay; oaz = az;

    // ---- projection: proj = points @ ABC -----------------------------------
    float msum = 0.f;
#pragma unroll
    for (int k = 0; k < PPT; ++k) {
#ifdef USE_WMMA44
      // Two 16x16x4 WMMAs per 32 points: B holds 16 points as columns (K=x,y,z,0),
      // A broadcasts (ax,ay,az,0) in every row -> D[m][n] = proj(point n).
      int s1 = lane & 15, s2 = s1 + 16;
      float x1 = __shfl(px[k], s1), y1 = __shfl(py[k], s1), z1 = __shfl(pz[k], s1);
      float x2 = __shfl(px[k], s2), y2 = __shfl(py[k], s2), z2 = __shfl(pz[k], s2);
      v2f Am;  Am.x = lolane ? ax : az;  Am.y = lolane ? ay : 0.f;
      v2f B1;  B1.x = lolane ? x1 : z1;  B1.y = lolane ? y1 : 0.f;
      v2f B2;  B2.x = lolane ? x2 : z2;  B2.y = lolane ? y2 : 0.f;
      v8f c0 = {0.f, 0.f, 0.f, 0.f, 0.f, 0.f, 0.f, 0.f};
      v8f d1 = __builtin_amdgcn_wmma_f32_16x16x4_f32(false, Am, false, B1,
                                                     (short)0, c0, false, false);
      v8f d2 = __builtin_amdgcn_wmma_f32_16x16x4_f32(false, Am, false, B2,
                                                     (short)0, c0, false, false);
      projv[k] = lolane ? d1[0] : d2[0];
#else
      projv[k] = ax * px[k] + ay * py[k] + az * pz[k];
#endif
      msum += projv[k] * mk[k];
    }
    const float mean = block_sum(msum, redf, tid) / (nsum + 1e-9f);

#pragma unroll
    for (int k = 0; k < PPT; ++k) {
      float s = projv[k] - mean;
      float sc = (s > 0.f) ? 1.f : ((s < 0.f) ? -1.f : 0.f);
      tot[k] += (int)sc * scale;
      ml[k] = fminf(fmaxf(-sc, 0.f), mk[k]);
      mr[k] = fminf(fmaxf( sc, 0.f), mk[k]);
    }

    // ---- duplet selection: 4 stable argmins of rand + (1-m)*1e9 ------------
    for (int q = 0; q < 4; ++q) {
      unsigned long long best = ~0ull;
#pragma unroll
      for (int k = 0; k < PPT; ++k) {
        int i = tid + k * T1;
        float m = (q < 2) ? ml[k] : mr[k];
        unsigned hs = hash_u32((unsigned)((((b * NDEPTH + d) * 4 + q) * NN) + i)
                               * 0x9E3779B1u + 0x7F4A7C15u);
        float u = u01(hs) + (1.f - m) * 1e9f;
        unsigned long long key =
            ((unsigned long long)__float_as_uint(u) << 32) | (unsigned)i;
        if (key < best) best = key;
      }
      unsigned long long w = block_min(best, redll, tid);
      int wi = (int)(w & 0xffffffffull);
      if (tid == (wi & (T1 - 1))) {
        int wk = wi >> 10;
        float sx = px[0], sy = py[0], sz = pz[0];
#pragma unroll
        for (int k = 1; k < PPT; ++k) {
          if (wk == k) { sx = px[k]; sy = py[k]; sz = pz[k]; }
        }
        dup[q][0] = sx; dup[q][1] = sy; dup[q][2] = sz;
      }
      __syncthreads();
    }
    float dlx = (dup[0][0] + dup[1][0]) * 0.5f;
    float dly = (dup[0][1] + dup[1][1]) * 0.5f;
    float dlz = (dup[0][2] + dup[1][2]) * 0.5f;
    float drx = (dup[2][0] + dup[3][0]) * 0.5f;
    float dry = (dup[2][1] + dup[3][1]) * 0.5f;
    float drz = (dup[2][2] + dup[3][2]) * 0.5f;

#pragma unroll
    for (int k = 0; k < PPT; ++k) {
      px[k] = (px[k] - dlx) * ml[k] + (px[k] - drx) * mr[k];
      py[k] = (py[k] - dly) * ml[k] + (py[k] - dry) * mr[k];
      pz[k] = (pz[k] - dlz) * ml[k] + (pz[k] - drz) * mr[k];
    }
  }

  // ---- stable argsort: 12-bit key + 13-bit index, in-LDS bitonic -----------
#pragma unroll
  for (int k = 0; k < PPT; ++k) {
    int i = tid + k * T1;
    int key = tot[k] + 1023 + ((mk[k] < 0.5f) ? 2048 : 0);  // inf = 2^11
    comp[i] = (key << 13) | i;
  }
  __syncthreads();
  for (int kk = 2; kk <= NN; kk <<= 1) {
    for (int j = kk >> 1; j > 0; j >>= 1) {
      for (int i = tid; i < NN; i += T1) {
        int ixj = i ^ j;
        if (ixj > i) {
          int a = comp[i], c = comp[ixj];
          bool up = ((i & kk) == 0);
          if ((a > c) == up) { comp[i] = c; comp[ixj] = a; }
        }
      }
      __syncthreads();
    }
  }
#pragma unroll
  for (int k = 0; k < PPT; ++k) {
    int r = tid + k * T1;
    int iv = comp[r] & (NN - 1);
    idx_ws[b * NN + r] = iv;                                        // for gather
    out[(size_t)BB * NN * DD + (size_t)b * NN + iv] = (float)r;     // rev
  }
}

// ============================================================================
// Kernel 2: the 256 MB permutation gather — done by the Tensor Data Mover.
// One wave builds a gather-mode D# (16 x 16-bit row indices, 16 KB tile),
// TENSOR_LOAD_TO_LDS pulls the rows, TENSOR_STORE_FROM_LDS writes them out
// contiguously. Zero VALU/VGPR data traffic; tracked on TENSORcnt.
// ============================================================================
__global__ __launch_bounds__(T2)
void gather_tdm_kernel(const float* __restrict__ x,
                       const int* __restrict__ idx_ws,
                       float* __restrict__ out) {
  __shared__ __align__(128) char stage[WAVES2][ROWS_PER_DESC * DD * 4];  // 2 x 16 KB

  const int wv = threadIdx.x >> 5;
  const int g = blockIdx.x * WAVES2 + wv;
  const int b = g / (NN / ROWS_PER_DESC);
  const int r0 = (g % (NN / ROWS_PER_DESC)) * ROWS_PER_DESC;
  const int* ip = idx_ws + b * NN + r0;

#ifdef USE_TDM
  // LDS byte address = low 32 bits of the flat shared-aperture address.
  unsigned lds = (unsigned)(uintptr_t)(&stage[wv][0]);
  unsigned long long src =
      (unsigned long long)(uintptr_t)(x + (size_t)b * NN * DD);
  unsigned long long dst =
      (unsigned long long)(uintptr_t)(out + ((size_t)b * NN + r0) * DD);

  // Pack 16 row indices as 16-bit pairs (gather_index_size = 0).
  int idxpk[8];
#pragma unroll
  for (int k = 0; k < 8; ++k) {
    unsigned a = (unsigned)ip[2 * k] & 0xffffu;
    unsigned c = (unsigned)ip[2 * k + 1] & 0xffffu;
    idxpk[k] = (int)(a | (c << 16));
  }

  // D# group 0: count=1, gather_mode=1 (load) / normal (store), type=2.
  v4u g0l = {1u | (1u << 31), lds,
             (unsigned)(src & 0xffffffffu),
             (unsigned)((src >> 32) & 0x01ffffffu) | (2u << 30)};
  v4u g0s = {1u, lds,
             (unsigned)(dst & 0xffffffffu),
             (unsigned)((dst >> 32) & 0x01ffffffu) | (2u << 30)};

  // D# group 1: data_size=4B, tensor 256 x 8192, tile 256 x 16, stride0=256.
  v8i g1 = {(int)(2u << 16),                               // data_size=2 (4B)
            (int)(((unsigned)DD & 0xffffu) << 16),         // tensor_dim0[15:0]
            (int)(((unsigned)DD >> 16) |
                  (((unsigned)NN & 0xffffu) << 16)),       // dim0 hi | dim1 lo
            (int)(((unsigned)NN >> 16) |
                  ((unsigned)DD << 16)),                   // dim1 hi | tile_dim0
            (int)ROWS_PER_DESC,                            // tile_dim1 (=#indices)
            (int)DD,                                       // tensor_dim0_stride lo
            0, 0};

  v4i g2 = {idxpk[0], idxpk[1], idxpk[2], idxpk[3]};
  v4i g3 = {idxpk[4], idxpk[5], idxpk[6], idxpk[7]};
  v4i z4 = {0, 0, 0, 0};
  v8i z8 = {0, 0, 0, 0, 0, 0, 0, 0};   // extra group of the 6-arg builtin (unused)

  // This toolchain exposes the 6-arg TDM builtin:
  //   (uint32x4 g0, int32x8 g1, int32x4 g2, int32x4 g3, int32x8 extra, i32 cpol)
  __builtin_amdgcn_tensor_load_to_lds(g0l, g1, g2, g3, z8, 0);
  __builtin_amdgcn_s_wait_tensorcnt(0);                    // load done before store reads LDS
  __builtin_amdgcn_tensor_store_from_lds(g0s, g1, z4, z4, z8, 0);
  __builtin_amdgcn_s_wait_tensorcnt(0);
#else
  // Fallback: direct per-wave row copy with 128-bit accesses.
  int lane = threadIdx.x & 31;
  for (int row = 0; row < ROWS_PER_DESC; ++row) {
    int srcr = ip[row];
    const float4* s = (const float4*)(x + ((size_t)b * NN + srcr) * DD);
    float4* dp = (float4*)(out + ((size_t)b * NN + r0 + row) * DD);
    for (int c = lane; c < DD / 4; c += 32) dp[c] = s[c];
  }
#endif
}

// ============================================================================
extern "C" void kernel_launch(void* const* d_in, const int* in_sizes, int n_in,
                              void* d_out, int out_size, void* d_ws, size_t ws_size,
                              hipStream_t stream) {
  const float* x      = (const float*)d_in[0];   // [B,N,256]
  const float* coords = (const float*)d_in[1];   // [B,N,3]
  const float* mask   = (const float*)d_in[2];   // [B,N]
  float* out = (float*)d_out;                    // [B,N,256] x_out ++ [B,N] rev
  int* idx_ws = (int*)d_ws;                      // [B,N] permutation (512 KB)

  serialize_kernel<<<BB, T1, 0, stream>>>(coords, mask, idx_ws, out);
  gather_tdm_kernel<<<NGROUP / WAVES2, T2, 0, stream>>>(x, idx_ws, out);
}